// MoraEncoder_51823075394081
// MI455X (gfx1250) — compile-verified
//
#include <hip/hip_runtime.h>
#include <hip/hip_bf16.h>
#include <stdint.h>

#ifndef __has_builtin
#define __has_builtin(x) 0
#endif

typedef __attribute__((ext_vector_type(4))) float v4f;

#define NMORAS  128
#define DMAX    20
#define ZBLOCK  256
#define ZUNROLL 8

// ---------------------------------------------------------------------------
// Kernel 1: bulk zero-fill of the whole output with NT b128 stores.
// Each block covers 2048 float4s; 8 stores share one address computation
// (k*4096-byte immediate offsets), so the inner loop is essentially pure
// global_store_b128 ... nt — what the 23.3 TB/s HBM path needs.
// ---------------------------------------------------------------------------
__global__ void zero_fill_kernel(v4f* __restrict__ out, long n4) {
    const long per_block  = (long)ZBLOCK * ZUNROLL;
    const long blockStart = (long)blockIdx.x * per_block;
    const long i          = blockStart + threadIdx.x;
    v4f z = {0.0f, 0.0f, 0.0f, 0.0f};
    if (blockStart + per_block <= n4) {
#pragma unroll
        for (int k = 0; k < ZUNROLL; ++k)
            __builtin_nontemporal_store(z, &out[i + (long)k * ZBLOCK]);
    } else {
        for (int k = 0; k < ZUNROLL; ++k) {
            long idx = i + (long)k * ZBLOCK;
            if (idx < n4) __builtin_nontemporal_store(z, &out[idx]);
        }
    }
}

__global__ void zero_tail_kernel(float* __restrict__ out, long start, long total) {
    long idx = start + (long)blockIdx.x * blockDim.x + threadIdx.x;
    if (idx < total) out[idx] = 0.0f;
}

// ---------------------------------------------------------------------------
// Kernel 2: event scatter. One thread per (event, offset<DMAX) pair; the 20
// threads of an event write a contiguous run so stores coalesce. Event
// metadata loads hit L0/L2 (20 lanes share the same addresses).
// Mirrors the reference's mode="drop" semantics.
// ---------------------------------------------------------------------------
__global__ void scatter_kernel(const float* __restrict__ pitches,
                               const int*   __restrict__ moras,
                               const int*   __restrict__ times,
                               const int*   __restrict__ durs,
                               const int*   __restrict__ pad_pre_ptr,
                               float*       __restrict__ out,
                               long T, int n_events, int t_sent) {
    long gid = (long)blockIdx.x * blockDim.x + threadIdx.x;
    int  e   = (int)(gid / DMAX);
    int  j   = (int)(gid - (long)e * DMAX);
    if (e >= n_events) return;
    if (j >= durs[e]) return;
    unsigned m = (unsigned)moras[e];
    if (m >= NMORAS) return;                        // drop OOB rows
    long scol = (long)times[e] + j;
    if (scol < 0 || scol >= (long)t_sent) return;   // drop OOB cols (pre-pad space)
    long col = (long)pad_pre_ptr[0] + scol;
    if (col < 0 || col >= T) return;
    out[(long)m * T + col] = pitches[e];
}

// ---------------------------------------------------------------------------
// Kernel 3: sound copy via the CDNA5 async global<->LDS path.
// Each lane moves 16 B: global_load_async_to_lds_b128, wait ASYNCcnt==0,
// global_store_async_from_lds_b128. No VGPR data movement at all.
// ---------------------------------------------------------------------------
__device__ __forceinline__ unsigned lds_off_u32(void* p) {
    return (unsigned)(unsigned long long)(__attribute__((address_space(3))) void*)p;
}

__global__ void sound_copy_kernel(const float* __restrict__ snd,
                                  float*       __restrict__ outsnd,
                                  const int*   __restrict__ pad_pre_ptr,
                                  int t_sent) {
    __shared__ v4f buf[ZBLOCK];
    const int tid = threadIdx.x;
    const int pp  = pad_pre_ptr[0];
    long base4 = ((long)blockIdx.x * ZBLOCK + tid) * 4;
    bool aligned = ((pp & 3) == 0);
    if (aligned && (base4 + 4 <= (long)t_sent)) {
        const float* gsrc = snd + base4;
        float*       gdst = outsnd + (long)pp + base4;
        unsigned     ldsa = lds_off_u32((void*)&buf[tid]);
#if __has_builtin(__builtin_amdgcn_s_wait_asynccnt)
        asm volatile("global_load_async_to_lds_b128 %0, %1, off"
                     :: "v"(ldsa), "v"(gsrc) : "memory");
        __builtin_amdgcn_s_wait_asynccnt(0);
        asm volatile("global_store_async_from_lds_b128 %0, %1, off"
                     :: "v"(gdst), "v"(ldsa) : "memory");
#else
        asm volatile("global_load_async_to_lds_b128 %0, %1, off"
                     :: "v"(ldsa), "v"(gsrc) : "memory");
        asm volatile("s_wait_asynccnt 0x0" ::: "memory");
        asm volatile("global_store_async_from_lds_b128 %0, %1, off"
                     :: "v"(gdst), "v"(ldsa) : "memory");
#endif
        // ASYNCcnt drains before wave end (implicit wait-idle at S_ENDPGM).
    } else {
        for (int k = 0; k < 4; ++k) {
            long idx = base4 + k;
            if (idx < (long)t_sent) outsnd[(long)pp + idx] = snd[idx];
        }
    }
}

// ---------------------------------------------------------------------------
extern "C" void kernel_launch(void* const* d_in, const int* in_sizes, int n_in,
                              void* d_out, int out_size, void* d_ws, size_t ws_size,
                              hipStream_t stream) {
    const float* pitches = (const float*)d_in[0];
    const float* sound   = (const float*)d_in[1];
    const int*   moras   = (const int*)d_in[2];
    const int*   times   = (const int*)d_in[3];
    const int*   durs    = (const int*)d_in[4];
    const int*   pad_pre = (const int*)d_in[5];
    // d_in[6] (pad_post) is implied by out_size.

    const int  n_events = in_sizes[0];
    const int  t_sent   = in_sizes[1];
    float*     out      = (float*)d_out;

    // out = [128 * T] moras ++ [T] sound  =>  T = out_size / 129
    const long T   = (long)out_size / (NMORAS + 1);
    const long n4  = (long)out_size / 4;
    const long rem = (long)out_size - n4 * 4;

    // 1) zero-fill everything (moras incl. pads, sound pads)
    const long per_block = (long)ZBLOCK * ZUNROLL;
    const long zblocks   = (n4 + per_block - 1) / per_block;
    if (zblocks > 0)
        zero_fill_kernel<<<(int)zblocks, ZBLOCK, 0, stream>>>((v4f*)out, n4);
    if (rem > 0)
        zero_tail_kernel<<<1, 4, 0, stream>>>(out, n4 * 4, (long)out_size);

    // 2) scatter event pitches into the mora matrix
    const long total   = (long)n_events * DMAX;
    const long sblocks = (total + 255) / 256;
    if (sblocks > 0)
        scatter_kernel<<<(int)sblocks, 256, 0, stream>>>(
            pitches, moras, times, durs, pad_pre, out, T, n_events, t_sent);

    // 3) copy the sound into the padded sound row
    float* outsnd = out + (long)NMORAS * T;
    const int cblocks = (int)(((long)t_sent + ZBLOCK * 4 - 1) / (ZBLOCK * 4));
    if (cblocks > 0)
        sound_copy_kernel<<<cblocks, ZBLOCK, 0, stream>>>(sound, outsnd, pad_pre, t_sent);
}